// DistanceNeighborSampler_90537910600155
// MI455X (gfx1250) — compile-verified
//
#include <hip/hip_runtime.h>
#include <stdint.h>

// ---------------------------------------------------------------------------
// DistanceNeighborSampler for MI455X (gfx1250, wave32)
//   d^2(x, n_k) = |x|^2 - 2 x.n_k + |n_k|^2  ; dots via V_WMMA_F32_16X16X4_F32
//   prob = normalize(exp(-sqrt(d^2 + 1e-12))), zeroed below EPS
//   categorical sampling: JAX threefry2x32 (key (0,42)) + Gumbel-max argmax
// ---------------------------------------------------------------------------

typedef float v2f __attribute__((ext_vector_type(2)));
typedef float v8f __attribute__((ext_vector_type(8)));

#define FEAT   128
#define NNEI   64
#define NSAMP  32
#define WPB    4     // waves per block (1 batch row per wave)

__device__ __forceinline__ uint32_t rotl32(uint32_t x, int n) {
  return (x << n) | (x >> (32 - n));
}

__device__ __forceinline__ void tf4(uint32_t& x0, uint32_t& x1,
                                    int r0, int r1, int r2, int r3) {
  x0 += x1; x1 = rotl32(x1, r0); x1 ^= x0;
  x0 += x1; x1 = rotl32(x1, r1); x1 ^= x0;
  x0 += x1; x1 = rotl32(x1, r2); x1 ^= x0;
  x0 += x1; x1 = rotl32(x1, r3); x1 ^= x0;
}

// bits[e] of jax threefry_random_bits(key=(0,42), n = 2*halfN):
//   e < halfN  -> out0 of block (e, e+halfN);  else out1 of block (e-halfN, e)
__device__ __forceinline__ float jax_gumbel(uint32_t e, uint32_t halfN) {
  const uint32_t k0 = 0u, k1 = 42u;
  const uint32_t k2 = 0x1BD11BDAu ^ k0 ^ k1;
  const bool lo = e < halfN;
  uint32_t x0 = lo ? e : (e - halfN);
  uint32_t x1 = lo ? (e + halfN) : e;
  x0 += k0; x1 += k1;
  tf4(x0, x1, 13, 15, 26,  6);  x0 += k1; x1 += k2 + 1u;
  tf4(x0, x1, 17, 29, 16, 24);  x0 += k2; x1 += k0 + 2u;
  tf4(x0, x1, 13, 15, 26,  6);  x0 += k0; x1 += k1 + 3u;
  tf4(x0, x1, 17, 29, 16, 24);  x0 += k1; x1 += k2 + 4u;
  tf4(x0, x1, 13, 15, 26,  6);  x0 += k2; x1 += k0 + 5u;
  const uint32_t bits = lo ? x0 : x1;
  // jax uniform(minval=tiny, maxval=1): mantissa trick into [1,2) - 1
  float f = __uint_as_float((bits >> 9) | 0x3f800000u) - 1.0f;
  const float TINY = 1.17549435e-38f;
  float u = fmaxf(f * (1.0f - TINY) + TINY, TINY);
  return -logf(-logf(u));
}

__global__ __launch_bounds__(WPB * 32)
void dns_kernel(const int* __restrict__ ids,
                const float* __restrict__ features,
                const int* __restrict__ adj,
                int* __restrict__ out,
                int batch, uint32_t halfN) {
  __shared__ float xbuf[WPB][FEAT];   // center features, per wave
  __shared__ int   abuf[WPB][NNEI];   // neighbor ids, per wave

  const int lane = threadIdx.x & 31;
  const int wave = threadIdx.x >> 5;
  const int b = blockIdx.x * WPB + wave;
  if (b >= batch) return;            // wave-uniform: EXEC stays full for WMMA

  const int idb = ids[b];

  // ---- center feature vector -> LDS, and |x|^2 --------------------------
  const float4 x4 =
      *reinterpret_cast<const float4*>(features + (size_t)idb * FEAT + lane * 4);
  float xn = x4.x * x4.x + x4.y * x4.y + x4.z * x4.z + x4.w * x4.w;
  #pragma unroll
  for (int m = 1; m < 32; m <<= 1) xn += __shfl_xor(xn, m, 32);
  *reinterpret_cast<float4*>(&xbuf[wave][lane * 4]) = x4;

  // ---- neighbor id row -> LDS, prefetch feature rows --------------------
  const int a0 = adj[(size_t)idb * NNEI + lane];
  const int a1 = adj[(size_t)idb * NNEI + 32 + lane];
  abuf[wave][lane]      = a0;
  abuf[wave][lane + 32] = a1;
  __builtin_prefetch(features + (size_t)a0 * FEAT, 0, 0);  // global_prefetch_b8
  __builtin_prefetch(features + (size_t)a1 * FEAT, 0, 0);
  __builtin_amdgcn_wave_barrier();   // same-wave DS ops are in-order in HW

  // ---- distances: 4 tiles of 16 neighbors, f32 WMMA over K-chunks of 4 --
  // Lane l owns neighbors k=l (slot 0) and k=l+32 (slot 1).
  float dot_own[2] = {0.f, 0.f};
  float nrm_own[2] = {0.f, 0.f};
  const int koff = (lane >= 16) ? 2 : 0;   // A/B 32-bit layout split

  #pragma unroll
  for (int t = 0; t < 4; ++t) {
    const int mloc = lane & 15;
    const int nid  = abuf[wave][t * 16 + mloc];
    const float* rowp = features + (size_t)nid * FEAT + koff;
    const float* xp   = &xbuf[wave][koff];

    v8f   acc = {};      // 16x16 f32 accumulator (all columns identical)
    float nrm = 0.f;     // partial |n|^2 (this lane's K slots)
    #pragma unroll 8
    for (int c = 0; c < FEAT; c += 4) {
      v2f a  = *reinterpret_cast<const v2f*>(rowp + c);  // A: neighbor chunk
      v2f bv = *reinterpret_cast<const v2f*>(xp + c);    // B: x replicated cols
      nrm += a.x * a.x + a.y * a.y;
      acc = __builtin_amdgcn_wmma_f32_16x16x4_f32(
          false, a, false, bv, (short)0, acc, false, false);
    }
    nrm += __shfl_xor(nrm, 16, 32);  // combine K mod4 {0,1} + {2,3} halves

    const int ownerHalf = t & 1;     // which lane half owns this tile's k's
    const int slot      = t >> 1;
    if ((lane >> 4) == ownerHalf) nrm_own[slot] = nrm;

    // D layout: row M=m in VGPR m&7, lanes (m>>3)*16..+15 (all cols equal)
    #pragma unroll
    for (int m = 0; m < 16; ++m) {
      float dv = __shfl(acc[m & 7], (m >> 3) << 4, 32);
      if (lane == ownerHalf * 16 + m) dot_own[slot] = dv;
    }
  }

  // ---- probabilities ----------------------------------------------------
  float p[2], logp[2];
  float psum = 0.f;
  #pragma unroll
  for (int s2 = 0; s2 < 2; ++s2) {
    float d2 = xn - 2.0f * dot_own[s2] + nrm_own[s2];
    float dist = sqrtf(fmaxf(d2, 0.0f) + 1e-12f);
    p[s2] = expf(-dist);
    psum += p[s2];
  }
  #pragma unroll
  for (int m = 1; m < 32; m <<= 1) psum += __shfl_xor(psum, m, 32);
  const float inv = 1.0f / psum;
  #pragma unroll
  for (int s2 = 0; s2 < 2; ++s2) {
    float pn = p[s2] * inv;
    logp[s2] = (pn > 1e-3f) ? logf(pn) : -__builtin_inff();
  }

  // ---- Gumbel-max categorical sampling ----------------------------------
  const uint32_t rowbase = ((uint32_t)b * NSAMP) * NNEI;
  for (int s = 0; s < NSAMP; ++s) {
    const uint32_t e0 = rowbase + (uint32_t)s * NNEI + (uint32_t)lane;
    const float g0 = jax_gumbel(e0, halfN);
    const float g1 = jax_gumbel(e0 + 32u, halfN);
    float s0 = logp[0] + g0;
    float s1 = logp[1] + g1;
    float bs; int bi;                       // lowest-index tie-break (argmax)
    if (s0 >= s1) { bs = s0; bi = lane; } else { bs = s1; bi = lane + 32; }
    #pragma unroll
    for (int m = 1; m < 32; m <<= 1) {
      float os = __shfl_xor(bs, m, 32);
      int   oi = __shfl_xor(bi, m, 32);
      if (os > bs || (os == bs && oi < bi)) { bs = os; bi = oi; }
    }
    if (lane == 0) out[(size_t)b * NSAMP + s] = abuf[wave][bi];
  }
}

extern "C" void kernel_launch(void* const* d_in, const int* in_sizes, int n_in,
                              void* d_out, int out_size, void* d_ws, size_t ws_size,
                              hipStream_t stream) {
  // setup_inputs order: ids, num_samples, features, batch_size, adj_info
  const int*   ids      = (const int*)d_in[0];
  const float* features = (const float*)d_in[2];
  const int*   adj      = (const int*)d_in[4];
  int*         out      = (int*)d_out;

  const int batch = in_sizes[0];
  const uint32_t halfN =
      (uint32_t)(((uint64_t)batch * (uint64_t)NSAMP * (uint64_t)NNEI) / 2);

  dim3 block(WPB * 32);
  dim3 grid((batch + WPB - 1) / WPB);
  hipLaunchKernelGGL(dns_kernel, grid, block, 0, stream,
                     ids, features, adj, out, batch, halfN);
}